// TreeLSTMCell_6648609374507
// MI455X (gfx1250) — compile-verified
//
#include <hip/hip_runtime.h>
#include <math.h>

// ---- problem sizes (match reference) ----
#define X_SIZE     256
#define H_SIZE     512
#define N_CHILDREN 32768

// ---- tiling ----
#define NB         128                  // children per workgroup
#define KC         64                   // K chunk staged in LDS
#define NCHUNK     (H_SIZE / KC)        // 8 K-chunks
#define NWG        (N_CHILDREN / NB)    // 256 workgroups in main kernel
#define LDS_STRIDE 68                   // 64 cols + pad 4 -> conflict-free b64 B reads

typedef float v2f  __attribute__((ext_vector_type(2)));
typedef float v4f  __attribute__((ext_vector_type(4)));
typedef float v8f  __attribute__((ext_vector_type(8)));

typedef unsigned int u32x4_t __attribute__((ext_vector_type(4)));
typedef int          i32x8_t __attribute__((ext_vector_type(8)));
typedef int          i32x4_t __attribute__((ext_vector_type(4)));

__device__ __forceinline__ float fast_sigmoid(float x) {
    return 1.0f / (1.0f + __expf(-x));
}
__device__ __forceinline__ float fast_tanh(float x) {
    float e = __expf(-2.0f * fabsf(x));
    float t = (1.0f - e) / (1.0f + e);
    return copysignf(t, x);
}

// Issue one TDM load: 2D tile KC x NB f32, tensor row stride H_SIZE f32,
// LDS padded 4 dwords every 64 dwords (-> row stride LDS_STRIDE dwords).
__device__ __forceinline__ void tdm_load_tile(unsigned int lds_byte_off,
                                              const float* gaddr) {
    unsigned long long ga = (unsigned long long)(uintptr_t)gaddr;
    u32x4_t g0;
    g0.x = 1u;                                    // count=1, user descriptor
    g0.y = lds_byte_off;                          // lds_addr (bytes)
    g0.z = (unsigned int)ga;                      // global_addr lo32
    g0.w = (unsigned int)((ga >> 32) & 0x01FFFFFFu) | (2u << 30);  // addr hi + type=2
    i32x8_t g1;
    // data_size=4B (2), pad_enable, pad_interval=64 dwords (5), pad_amount=4 dwords (3)
    g1[0] = (int)((2u << 16) | (1u << 20) | (5u << 22) | (3u << 25));
    g1[1] = (int)(((unsigned)H_SIZE & 0xFFFFu) << 16);      // tensor_dim0 = 512 (lo16)
    g1[2] = (int)(((unsigned)N_CHILDREN & 0xFFFFu) << 16);  // dim0 hi=0 | tensor_dim1 lo16
    g1[3] = (int)((unsigned)KC << 16);                      // dim1 hi=0 | tile_dim0 = 64
    g1[4] = NB;                                             // tile_dim1 = 128, tile_dim2 = 0
    g1[5] = H_SIZE;                                         // tensor_dim0_stride = 512 (lo32)
    g1[6] = 0;
    g1[7] = 0;
    i32x4_t gz4 = {0, 0, 0, 0};
    i32x8_t gz8 = {0, 0, 0, 0, 0, 0, 0, 0};
    __builtin_amdgcn_tensor_load_to_lds(g0, g1, gz4, gz4, gz8, 0);
}

// ---------------- kernel 0: wfxb[h] = w_f[h,:].x + b_f[h] ----------------
__global__ __launch_bounds__(128)
void k_wfx(const float* __restrict__ x, const float* __restrict__ w_f,
           const float* __restrict__ b_f, float* __restrict__ wfxb) {
    __shared__ float xs[X_SIZE];
    int t = threadIdx.x;
    for (int i = t; i < X_SIZE; i += 128) xs[i] = x[i];
    __syncthreads();
    int h = blockIdx.x * 128 + t;
    float s = b_f[h];
    const float* wr = w_f + (size_t)h * X_SIZE;
    #pragma unroll 4
    for (int j = 0; j < X_SIZE; ++j) s += wr[j] * xs[j];
    wfxb[h] = s;
}

// ---------------- kernel 1: main fused GEMM (f32 WMMA + TDM staging) ----------------
// fs[m,n] = sum_k u_f[m,k] * hs[cb+n,k]  -> sigmoid(fs + wfxb[m]) * cs[cb+n,m]
// -> per-WG partial c contribution + per-WG partial h_sum.
__global__ __launch_bounds__(512)
void k_main(const float* __restrict__ hs, const float* __restrict__ cs,
            const float* __restrict__ u_f, const float* __restrict__ wfxb,
            float* __restrict__ hsum_part, float* __restrict__ c_part) {
    __shared__ float lds[2][NB * LDS_STRIDE];   // double-buffered hs chunk, row stride 68

    const int t     = threadIdx.x;
    const int wave  = t >> 5;                // 0..15
    const int lane  = t & 31;
    const int l16   = lane & 15;
    const int hi    = lane >> 4;             // lane-half (K offset +2 per ISA A/B layout)
    const int cb    = blockIdx.x * NB;       // child base
    const int mbase = wave * 32;             // this wave's 32 h-rows (2 M-tiles)

    // wave 0 kicks off the first tile DMA
    if (wave == 0) {
        tdm_load_tile((unsigned int)(uintptr_t)&lds[0][0],
                      hs + (size_t)cb * H_SIZE);
    }

    v8f acc[2][8];
    #pragma unroll
    for (int mt = 0; mt < 2; ++mt)
        #pragma unroll
        for (int nt = 0; nt < 8; ++nt)
            acc[mt][nt] = (v8f)0.0f;

    for (int i = 0; i < NCHUNK; ++i) {
        const int k0 = i * KC;
        // chunk i's DMA (issued last iteration or in prologue) must be complete
        if (wave == 0) {
            __builtin_amdgcn_s_wait_tensorcnt(0);
        }
        __syncthreads();   // chunk i visible to all; prior reads of buf[(i+1)&1] done
        // prefetch chunk i+1 into the other buffer (overlaps WMMA sweep below)
        if (wave == 0 && (i + 1) < NCHUNK) {
            tdm_load_tile((unsigned int)(uintptr_t)&lds[(i + 1) & 1][0],
                          hs + (size_t)cb * H_SIZE + (k0 + KC));
        }
        const float* buf = &lds[i & 1][0];

        // ---- per-WG h_sum partial for this K slice (column sums over children) ----
        if (t < KC) {
            float s = 0.0f;
            #pragma unroll 8
            for (int n = 0; n < NB; ++n) s += buf[n * LDS_STRIDE + t];
            hsum_part[(size_t)blockIdx.x * H_SIZE + k0 + t] = s;
        }

        // ---- WMMA sweep: 16 K-steps x (2 M-tiles x 8 N-tiles) ----
        #pragma unroll 4
        for (int kk = 0; kk < KC; kk += 4) {
            const int kg = k0 + kk + hi * 2;
            // A tiles from u_f (ISA 16x4 f32 A layout: lane m=l16, K = 2*half + vgpr)
            v2f a0 = *(const v2f*)(u_f + (size_t)(mbase +      l16) * H_SIZE + kg);
            v2f a1 = *(const v2f*)(u_f + (size_t)(mbase + 16 + l16) * H_SIZE + kg);
            #pragma unroll
            for (int nt = 0; nt < 8; ++nt) {
                // B[k][n] = hs[n][k]: lane n=l16, K = 2*half + vgpr
                const int n = nt * 16 + l16;
                v2f b = *(const v2f*)(&buf[n * LDS_STRIDE + kk + hi * 2]);
                acc[0][nt] = __builtin_amdgcn_wmma_f32_16x16x4_f32(
                    false, a0, false, b, (short)0, acc[0][nt], false, false);
                acc[1][nt] = __builtin_amdgcn_wmma_f32_16x16x4_f32(
                    false, a1, false, b, (short)0, acc[1][nt], false, false);
            }
        }
    }

    // ---- epilogue: sigmoid + cs-weighted reduction over children ----
    // C/D layout: lane L, vgpr r holds fs[m = tile_m + r + 8*hi][n = tile_n + l16]
    #pragma unroll
    for (int mt = 0; mt < 2; ++mt) {
        const int mrow = mbase + mt * 16 + hi * 8;   // 8 consecutive m per lane
        float addv[8];
        #pragma unroll
        for (int r = 0; r < 8; ++r) addv[r] = wfxb[mrow + r];

        float csum[8];
        #pragma unroll
        for (int r = 0; r < 8; ++r) csum[r] = 0.0f;

        #pragma unroll
        for (int nt = 0; nt < 8; ++nt) {
            const int n = cb + nt * 16 + l16;
            v4f c0 = *(const v4f*)(cs + (size_t)n * H_SIZE + mrow);
            v4f c1 = *(const v4f*)(cs + (size_t)n * H_SIZE + mrow + 4);
            #pragma unroll
            for (int r = 0; r < 8; ++r) {
                float f  = fast_sigmoid(acc[mt][nt][r] + addv[r]);
                float cv = (r < 4) ? c0[r] : c1[r - 4];
                csum[r] += f * cv;
            }
        }
        // reduce over the 16 lanes sharing the same m (xor within 16-lane half)
        #pragma unroll
        for (int r = 0; r < 8; ++r) {
            float v = csum[r];
            v += __shfl_xor(v, 1, 32);
            v += __shfl_xor(v, 2, 32);
            v += __shfl_xor(v, 4, 32);
            v += __shfl_xor(v, 8, 32);
            csum[r] = v;
        }
        if (l16 == 0) {
            #pragma unroll
            for (int r = 0; r < 8; ++r)
                c_part[(size_t)blockIdx.x * H_SIZE + mrow + r] = csum[r];
        }
    }
}

// ---------------- kernel 2: reduce per-WG partials ----------------
__global__ __launch_bounds__(256)
void k_reduce(const float* __restrict__ hsum_part, const float* __restrict__ c_part,
              float* __restrict__ hsum, float* __restrict__ c_extra) {
    int g = blockIdx.x * 256 + threadIdx.x;     // 0..1023
    const float* src = (g < H_SIZE) ? hsum_part : c_part;
    int h = (g < H_SIZE) ? g : g - H_SIZE;
    float s = 0.0f;
    #pragma unroll 8
    for (int w = 0; w < NWG; ++w) s += src[(size_t)w * H_SIZE + h];
    if (g < H_SIZE) hsum[h] = s; else c_extra[h] = s;
}

// ---------------- kernel 3: iou gates (3x512 GEMV over 12 WGPs) ----------------
__global__ __launch_bounds__(128)
void k_iou(const float* __restrict__ x, const float* __restrict__ w_iou,
           const float* __restrict__ u_iou, const float* __restrict__ b_iou,
           const float* __restrict__ hsum, float* __restrict__ iou_out) {
    __shared__ float xs[X_SIZE];
    __shared__ float hl[H_SIZE];
    int t = threadIdx.x;
    for (int i = t; i < X_SIZE; i += 128) xs[i] = x[i];
    for (int i = t; i < H_SIZE; i += 128) hl[i] = hsum[i];
    __syncthreads();
    int g = blockIdx.x * 128 + t;               // 0..1535 == k*512 + h
    float s = b_iou[g];
    const float* wr = w_iou + (size_t)g * X_SIZE;
    #pragma unroll 4
    for (int j = 0; j < X_SIZE; ++j) s += wr[j] * xs[j];
    const float* ur = u_iou + (size_t)g * H_SIZE;
    #pragma unroll 4
    for (int j = 0; j < H_SIZE; ++j) s += ur[j] * hl[j];
    iou_out[g] = s;
}

// ---------------- kernel 4: final combine ----------------
__global__ __launch_bounds__(512)
void k_final(const float* __restrict__ iou, const float* __restrict__ c_extra,
             float* __restrict__ out) {
    int h = threadIdx.x;
    float i = fast_sigmoid(iou[h]);
    float o = fast_sigmoid(iou[H_SIZE + h]);
    float u = fast_tanh(iou[2 * H_SIZE + h]);
    float c = i * u + c_extra[h];
    out[h]          = o * fast_tanh(c);   // h output
    out[H_SIZE + h] = c;                  // c output
}

extern "C" void kernel_launch(void* const* d_in, const int* in_sizes, int n_in,
                              void* d_out, int out_size, void* d_ws, size_t ws_size,
                              hipStream_t stream) {
    (void)in_sizes; (void)n_in; (void)out_size; (void)ws_size;
    const float* x     = (const float*)d_in[0];
    const float* hs    = (const float*)d_in[1];
    const float* cs    = (const float*)d_in[2];
    const float* w_iou = (const float*)d_in[3];
    const float* u_iou = (const float*)d_in[4];
    const float* b_iou = (const float*)d_in[5];
    const float* w_f   = (const float*)d_in[6];
    const float* u_f   = (const float*)d_in[7];
    const float* b_f   = (const float*)d_in[8];
    float* out = (float*)d_out;

    // workspace layout (floats)
    float* ws        = (float*)d_ws;
    float* wfxb      = ws;                                 // 512
    float* hsum_part = wfxb + H_SIZE;                      // NWG*512
    float* c_part    = hsum_part + (size_t)NWG * H_SIZE;   // NWG*512
    float* hsum      = c_part + (size_t)NWG * H_SIZE;      // 512
    float* c_extra   = hsum + H_SIZE;                      // 512
    float* iou       = c_extra + H_SIZE;                   // 1536

    k_wfx   <<<4,   128, 0, stream>>>(x, w_f, b_f, wfxb);
    k_main  <<<NWG, 512, 0, stream>>>(hs, cs, u_f, wfxb, hsum_part, c_part);
    k_reduce<<<4,   256, 0, stream>>>(hsum_part, c_part, hsum, c_extra);
    k_iou   <<<12,  128, 0, stream>>>(x, w_iou, u_iou, b_iou, hsum, iou);
    k_final <<<1,   512, 0, stream>>>(iou, c_extra, out);
}